// AdaptiveMetaLearnerV1_66692252172666
// MI455X (gfx1250) — compile-verified
//
#include <hip/hip_runtime.h>
#include <math.h>

// Problem constants (match reference)
#define P_DIM 10000
#define B_DIM 64
#define NROWS (P_DIM * B_DIM)   // 640000 rows
#define H 40
#define G 160                   // 4*H gates
#define EPSV 1e-5f
#define THREADS 256
#define NBLOCKS (NROWS / THREADS) // 2500, exact

// workspace layout (floats)
#define BR_STRIDE 656
#define A_OFF   0
#define BB_OFF  160
#define CH0_OFF 320
#define CH1_OFF 480
#define ST_OFF  640
#define QSUM_OFF 1536           // + NBLOCKS floats; total ws use ~16 KB

typedef float v2f __attribute__((ext_vector_type(2)));
typedef float v8f __attribute__((ext_vector_type(8)));

#if __has_builtin(__builtin_amdgcn_tanhf)
#define TANHF(x) __builtin_amdgcn_tanhf(x)
#else
#define TANHF(x) tanhf(x)
#endif

__device__ __forceinline__ float sigf(float x) {
  return 0.5f * TANHF(0.5f * x) + 0.5f;   // sigmoid via HW tanh
}

struct AP {
  const float *x, *W1, *b1, *Wih, *bih, *bhh, *g_ih, *be_ih, *g_hh, *be_hh,
              *g_c, *be_c, *Wout, *bout;
  const float *aW1, *ab1, *aWih, *abih, *abhh, *ag_ih, *abe_ih, *ag_hh,
              *abe_hh, *ag_c, *abe_c, *lambda_q, *aWout, *about;
  float *out;
  float *ws;
};

// -------- setup: rank-1 layer-0 folding + LN(bhh) constants + stats --------
__global__ void aml_setup_kernel(AP p) {
  int t = threadIdx.x;          // 0..319
  int b = t / G, j = t % G;
  const float* W1v  = b ? p.aW1   : p.W1;
  const float* b1v  = b ? p.ab1   : p.b1;
  const float* Wih  = b ? p.aWih  : p.Wih;
  const float* bih  = b ? p.abih  : p.bih;
  const float* bhh  = b ? p.abhh  : p.bhh;
  const float* ghh  = b ? p.ag_hh : p.g_hh;
  const float* behh = b ? p.abe_hh: p.be_hh;
  float* wsb = p.ws + b * BR_STRIDE;

  __shared__ float sa[2][G], sb[2][G];

  float a = 0.f, bb = 0.f;
  for (int k = 0; k < H; ++k) {
    float w = Wih[j * H + k];   // layer 0 weights
    a  += w * W1v[k];
    bb += w * b1v[k];
  }
  bb += bih[j];
  wsb[A_OFF + j] = a;
  wsb[BB_OFF + j] = bb;
  sa[b][j] = a; sb[b][j] = bb;

  // LN(bhh_l)*g_hh+be_hh per layer (redundant serial mean/var: trivial)
  for (int l = 0; l < 2; ++l) {
    float s = 0.f, q = 0.f;
    for (int i = 0; i < G; ++i) { float v = bhh[l * G + i]; s += v; q += v * v; }
    float m = s / (float)G;
    float rs = rsqrtf(q / (float)G - m * m + EPSV);
    wsb[(l ? CH1_OFF : CH0_OFF) + j] =
        (bhh[l * G + j] - m) * rs * ghh[l * G + j] + behh[l * G + j];
  }
  __syncthreads();
  if (j == 0) {                 // closed-form LN stats of x*a+b
    float sA = 0, sB = 0, sAA = 0, sAB = 0, sBB = 0;
    for (int i = 0; i < G; ++i) {
      float av = sa[b][i], bv = sb[b][i];
      sA += av; sB += bv; sAA += av * av; sAB += av * bv; sBB += bv * bv;
    }
    float ma = sA / (float)G, mb = sB / (float)G;
    wsb[ST_OFF + 0] = ma;
    wsb[ST_OFF + 1] = mb;
    wsb[ST_OFF + 2] = sAA / (float)G - ma * ma;
    wsb[ST_OFF + 3] = sAB / (float)G - ma * mb;
    wsb[ST_OFF + 4] = sBB / (float)G - mb * mb;
  }
}

// ------------------------------- main kernel -------------------------------
__global__ void __launch_bounds__(THREADS)
aml_main_kernel(AP p) {
  __shared__ float hbuf[8][32][41];   // per-wave h1 tiles, padded stride
  __shared__ float qsh[8];

  const int lane = threadIdx.x & 31;
  const int wave = threadIdx.x >> 5;
  const int c    = lane & 15;         // column within 16-lane half
  const int half = lane >> 4;
  const int rowBase = blockIdx.x * THREADS + wave * 32;
  const float xv = p.x[rowBase + lane];
  const float lam = p.lambda_q[0];
  float qacc = 0.f;

  for (int br = 0; br < 2; ++br) {
    const float* ws    = p.ws + br * BR_STRIDE;
    const float* g_ih  = br ? p.ag_ih  : p.g_ih;
    const float* be_ih = br ? p.abe_ih : p.be_ih;
    const float* bih   = br ? p.abih   : p.bih;
    const float* g_c   = br ? p.ag_c   : p.g_c;
    const float* be_c  = br ? p.abe_c  : p.be_c;
    const float* Wih1  = (br ? p.aWih : p.Wih) + G * H;  // layer-1 weights
    const float* Wout  = br ? p.aWout : p.Wout;
    const float  outb  = br ? p.about[0] : p.bout[0];

    // ---- layer 0: rank-1 closed form, per-lane row ----
    const float ma = ws[ST_OFF + 0], mb = ws[ST_OFF + 1];
    const float vaa = ws[ST_OFF + 2], vab = ws[ST_OFF + 3], vbb = ws[ST_OFF + 4];
    const float m0  = xv * ma + mb;
    const float rs0 = rsqrtf(xv * xv * vaa + 2.f * xv * vab + vbb + EPSV);

    float cs = 0.f, cq = 0.f;
    for (int k = 0; k < H; ++k) {          // f-gate skipped (c0 == 0)
      float iraw = (xv * ws[A_OFF + k] + ws[BB_OFF + k] - m0) * rs0 * g_ih[k]
                 + be_ih[k] + ws[CH0_OFF + k];
      int kg = 2 * H + k;
      float graw = (xv * ws[A_OFF + kg] + ws[BB_OFF + kg] - m0) * rs0 * g_ih[kg]
                 + be_ih[kg] + ws[CH0_OFF + kg];
      float cv = sigf(iraw) * TANHF(graw);
      cs += cv; cq += cv * cv;
      hbuf[wave][lane][k] = cv;
    }
    float mc  = cs / (float)H;
    float rsc = rsqrtf(cq / (float)H - mc * mc + EPSV);
    for (int k = 0; k < H; ++k) {
      int ko = 3 * H + k;
      float oraw = (xv * ws[A_OFF + ko] + ws[BB_OFF + ko] - m0) * rs0 * g_ih[ko]
                 + be_ih[ko] + ws[CH0_OFF + ko];
      float cv = hbuf[wave][lane][k];
      hbuf[wave][lane][k] =
          sigf(oraw) * TANHF((cv - mc) * rsc * g_c[k] + be_c[k]);
    }
    __syncthreads();   // h1 visible for cross-lane A-fragment gathers

    // per-lane layer-1 gate constants at gate = 16j + c
    float gi1[10], bi1[10], bj1[10], cj1[10];
    #pragma unroll
    for (int j = 0; j < 10; ++j) {
      int gg = 16 * j + c;
      gi1[j] = g_ih[G + gg];
      bi1[j] = be_ih[G + gg];
      bj1[j] = bih[G + gg];
      cj1[j] = ws[CH1_OFF + gg];
    }

    // ---- layer 1: fp32 WMMA  h1[16x40] @ Wih1^T[40x160] ----
    for (int mt = 0; mt < 2; ++mt) {
      v8f acc[10];
      #pragma unroll
      for (int j = 0; j < 10; ++j) acc[j] = (v8f){0,0,0,0,0,0,0,0};

      const int srcRow = mt * 16 + c;
      #pragma unroll
      for (int kt = 0; kt < 10; ++kt) {
        const int k0 = 4 * kt + 2 * half;       // A: lanes0-15 K,K+1; 16-31 K+2,K+3
        v2f Af; Af.x = hbuf[wave][srcRow][k0];
                Af.y = hbuf[wave][srcRow][k0 + 1];
        #pragma unroll
        for (int j = 0; j < 10; ++j) {
          const int gidx = 16 * j + c;
          v2f Bf; Bf.x = Wih1[gidx * H + k0];
                  Bf.y = Wih1[gidx * H + k0 + 1];
          acc[j] = __builtin_amdgcn_wmma_f32_16x16x4_f32(
              false, Af, false, Bf, (short)0, acc[j], false, false);
        }
      }

      // add bih1; LayerNorm stats over 160 gates per row (half-wave reduce)
      float s[8], q[8];
      #pragma unroll
      for (int v = 0; v < 8; ++v) { s[v] = 0.f; q[v] = 0.f; }
      #pragma unroll
      for (int j = 0; j < 10; ++j) {
        #pragma unroll
        for (int v = 0; v < 8; ++v) {
          float z = acc[j][v] + bj1[j];
          acc[j][v] = z;
          s[v] += z; q[v] += z * z;
        }
      }
      #pragma unroll
      for (int v = 0; v < 8; ++v)
        for (int off = 1; off < 16; off <<= 1) {
          s[v] += __shfl_xor(s[v], off);
          q[v] += __shfl_xor(q[v], off);
        }

      #pragma unroll
      for (int v = 0; v < 8; ++v) {
        float mean = s[v] / (float)G;
        float rs   = rsqrtf(q[v] / (float)G - mean * mean + EPSV);
        float gn[10];
        #pragma unroll
        for (int j = 0; j < 10; ++j)
          gn[j] = (acc[j][v] - mean) * rs * gi1[j] + bi1[j] + cj1[j];

        // c = sig(i)*tanh(g); i at tile kg, g at tile 5+kg (80 = 5*16 aligned)
        float cv0 = sigf(gn[0]) * TANHF(gn[5]);
        float cv1 = sigf(gn[1]) * TANHF(gn[6]);
        float cv2 = sigf(gn[2]) * TANHF(gn[7]);      // valid for c<8 only
        float csm = cv0 + cv1 + ((c < 8) ? cv2 : 0.f);
        float cqm = cv0*cv0 + cv1*cv1 + ((c < 8) ? cv2*cv2 : 0.f);
        for (int off = 1; off < 16; off <<= 1) {
          csm += __shfl_xor(csm, off);
          cqm += __shfl_xor(cqm, off);
        }
        float mcc = csm / (float)H;
        float rcc = rsqrtf(cqm / (float)H - mcc * mcc + EPSV);

        // o-gate (offset 120 = 7.5 tiles): fix 8-column skew with shfl_xor 8
        float t0 = (c >= 8) ? gn[7] : gn[8];
        float t1 = (c >= 8) ? gn[8] : gn[9];
        float t2 = gn[9];
        float so0 = sigf(__shfl_xor(t0, 8));
        float so1 = sigf(__shfl_xor(t1, 8));
        float so2 = sigf(__shfl_xor(t2, 8));

        float h0 = so0 * TANHF((cv0 - mcc) * rcc * g_c[H + c]      + be_c[H + c]);
        float h1 = so1 * TANHF((cv1 - mcc) * rcc * g_c[H + 16 + c] + be_c[H + 16 + c]);
        float po = h0 * Wout[c] + h1 * Wout[16 + c];
        if (c < 8) {
          float h2 = so2 * TANHF((cv2 - mcc) * rcc * g_c[H + 32 + c] + be_c[H + 32 + c]);
          po += h2 * Wout[32 + c];
        }
        for (int off = 1; off < 16; off <<= 1) po += __shfl_xor(po, off);

        float val = po + outb;
        if (c == v) {
          int row = rowBase + mt * 16 + v + 8 * half;
          if (br == 0) p.out[row] = val;
          else         qacc += lam * TANHF(val);
        }
      }
    } // mt
    __syncthreads();   // protect hbuf before next branch reuses it
  } // br

  // deterministic per-block q partial sum
  for (int off = 1; off < 32; off <<= 1) qacc += __shfl_xor(qacc, off);
  if (lane == 0) qsh[wave] = qacc;
  __syncthreads();
  if (threadIdx.x == 0) {
    float bsum = 0.f;
    for (int w = 0; w < 8; ++w) bsum += qsh[w];
    p.ws[QSUM_OFF + blockIdx.x] = bsum;
  }
}

// ---------------- deterministic final mean reduction ----------------
__global__ void aml_reduce_kernel(AP p) {
  __shared__ float sm[256];
  int t = threadIdx.x;
  float s = 0.f;
  for (int i = t; i < NBLOCKS; i += 256) s += p.ws[QSUM_OFF + i];
  sm[t] = s;
  __syncthreads();
  for (int st = 128; st > 0; st >>= 1) {
    if (t < st) sm[t] += sm[t + st];
    __syncthreads();
  }
  if (t == 0) p.out[NROWS] = sm[0] / (float)NROWS;
}

extern "C" void kernel_launch(void* const* d_in, const int* in_sizes, int n_in,
                              void* d_out, int out_size, void* d_ws, size_t ws_size,
                              hipStream_t stream) {
  AP p;
  p.x      = (const float*)d_in[0];
  p.W1     = (const float*)d_in[1];
  p.b1     = (const float*)d_in[2];
  p.Wih    = (const float*)d_in[3];
  // d_in[4] = Whh (unused: h0 == 0)
  p.bih    = (const float*)d_in[5];
  p.bhh    = (const float*)d_in[6];
  p.g_ih   = (const float*)d_in[7];
  p.be_ih  = (const float*)d_in[8];
  p.g_hh   = (const float*)d_in[9];
  p.be_hh  = (const float*)d_in[10];
  p.g_c    = (const float*)d_in[11];
  p.be_c   = (const float*)d_in[12];
  p.Wout   = (const float*)d_in[13];
  p.bout   = (const float*)d_in[14];
  p.aW1    = (const float*)d_in[15];
  p.ab1    = (const float*)d_in[16];
  p.aWih   = (const float*)d_in[17];
  // d_in[18] = aWhh (unused)
  p.abih   = (const float*)d_in[19];
  p.abhh   = (const float*)d_in[20];
  p.ag_ih  = (const float*)d_in[21];
  p.abe_ih = (const float*)d_in[22];
  p.ag_hh  = (const float*)d_in[23];
  p.abe_hh = (const float*)d_in[24];
  p.ag_c   = (const float*)d_in[25];
  p.abe_c  = (const float*)d_in[26];
  p.lambda_q = (const float*)d_in[27];
  p.aWout  = (const float*)d_in[28];
  p.about  = (const float*)d_in[29];
  p.out    = (float*)d_out;
  p.ws     = (float*)d_ws;

  aml_setup_kernel<<<1, 320, 0, stream>>>(p);
  aml_main_kernel<<<NBLOCKS, THREADS, 0, stream>>>(p);
  aml_reduce_kernel<<<1, 256, 0, stream>>>(p);
}